// Forward_Model_27616639714079
// MI455X (gfx1250) — compile-verified
//
#include <hip/hip_runtime.h>
#include <math.h>

typedef float v2f __attribute__((ext_vector_type(2)));
typedef float v8f __attribute__((ext_vector_type(8)));

#define IMG 12
#define MATF 144                 // 12*12 floats per matrix
#define LDS_STRIDE 17            // 16-row tile, padded pitch (breaks M / M+8 bank aliasing)
#define TILE_F (16 * LDS_STRIDE) // 272 floats per staged tile
#define WAVES_PER_BLOCK 4
#define BLOCK_THREADS (WAVES_PER_BLOCK * 32)

// out = |Wq X Wq| / 65536^2  with Wq = round(W * 65535), W = 12x12 DFT real/imag parts.
// Forward value of the reference is exactly the "approx" pass (stop_gradient identity).
// One wave <-> one 12x12 matrix, zero-padded into a 16x16 WMMA tile.
__global__ __launch_bounds__(BLOCK_THREADS) void dft12_wmma_kernel(
    const float* __restrict__ x,   // (nMat, 12, 12)
    const float* __restrict__ wr,  // (144,)
    const float* __restrict__ wi,  // (144,)
    float* __restrict__ out,       // (nMat, 12, 12)
    int nMat)
{
    __shared__ float lds[WAVES_PER_BLOCK * 3 * TILE_F];

    const int lane = threadIdx.x & 31;
    const int wave = threadIdx.x >> 5;
    const int col  = lane & 15;          // N (or M for A operands)
    const int hi   = (lane >> 4) & 1;    // lane half selects K+2 / M+8
    const int koff = hi * 2;

    // branch-free padding: clamp address, select zero
    const bool colOk = (col < IMG);
    const int  colC  = colOk ? col : (IMG - 1);

    float* ldsTr = &lds[wave * 3 * TILE_F];
    float* ldsTi = ldsTr + TILE_F;
    float* ldsMg = ldsTi + TILE_F;

    // ---- build quantized weights in WMMA A-layout (and transposed / negated copies)
    // A layout (16x4 f32): reg j, lane l: element W[l%16][4c + j + 2*(l>=16)]
    v2f rwA[3], iwA[3], rwT[3], iwT[3], niT[3];
#pragma unroll
    for (int c = 0; c < 3; ++c) {
#pragma unroll
        for (int j = 0; j < 2; ++j) {
            const int k = 4 * c + j + koff;      // 0..11 always
            float ar = rintf(wr[colC * IMG + k] * 65535.0f);   // Wr[row][k]
            float ai = rintf(wi[colC * IMG + k] * 65535.0f);   // Wi[row][k]
            float tr = rintf(wr[k * IMG + colC] * 65535.0f);   // Wr^T[row][k] = Wr[k][row]
            float ti = rintf(wi[k * IMG + colC] * 65535.0f);
            rwA[c][j] = colOk ? ar : 0.f;
            iwA[c][j] = colOk ? ai : 0.f;
            rwT[c][j] = colOk ? tr : 0.f;
            iwT[c][j] = colOk ? ti : 0.f;
            niT[c][j] = colOk ? -ti : 0.f;
        }
    }

    const int matStride = gridDim.x * WAVES_PER_BLOCK;
    for (int base = blockIdx.x * WAVES_PER_BLOCK; base < nMat; base += matStride) {
        const int  mat    = base + wave;
        const bool active = (mat < nMat);
        const int  matC   = active ? mat : (nMat - 1);   // always-valid address
        const float* xm = x + (size_t)matC * MATF;

        // ---- stage 1: Tr = Wr_q @ X, Ti = Wi_q @ X   (K=12 as 3 chained K=4 WMMAs)
        v8f Tr = {}, Ti = {};
#pragma unroll
        for (int c = 0; c < 3; ++c) {
            v2f Xb;
#pragma unroll
            for (int j = 0; j < 2; ++j) {
                const int k = 4 * c + j + koff;          // row of X, 0..11
                const float v = xm[k * IMG + colC];      // unconditional load
                Xb[j] = colOk ? v : 0.f;                 // v_cndmask, no EXEC branch
            }
            Tr = __builtin_amdgcn_wmma_f32_16x16x4_f32(false, rwA[c], false, Xb,
                                                       (short)0, Tr, false, false);
            Ti = __builtin_amdgcn_wmma_f32_16x16x4_f32(false, iwA[c], false, Xb,
                                                       (short)0, Ti, false, false);
        }

        __syncthreads();  // WAR vs previous iteration's LDS reads
        // C-layout -> LDS: VGPR v holds row M = v + 8*hi, column N = col
#pragma unroll
        for (int v = 0; v < 8; ++v) {
            const int m = v + hi * 8;
            ldsTr[m * LDS_STRIDE + col] = Tr[v];
            ldsTi[m * LDS_STRIDE + col] = Ti[v];
        }
        __syncthreads();  // RAW: Tr/Ti visible across lanes

        // ---- stage 2: Or^T = Wr^T Tr^T - Wi^T Ti^T ; Oi^T = Wi^T Tr^T + Wr^T Ti^T
        v8f OrT = {}, OiT = {};
#pragma unroll
        for (int c = 0; c < 3; ++c) {
            v2f Br, Bi;
#pragma unroll
            for (int j = 0; j < 2; ++j) {
                const int k = 4 * c + j + koff;          // K row of T^T = column of T
                Br[j] = ldsTr[col * LDS_STRIDE + k];     // T^T[k][col] = T[col][k]
                Bi[j] = ldsTi[col * LDS_STRIDE + k];
            }
            OrT = __builtin_amdgcn_wmma_f32_16x16x4_f32(false, rwT[c], false, Br,
                                                        (short)0, OrT, false, false);
            OrT = __builtin_amdgcn_wmma_f32_16x16x4_f32(false, niT[c], false, Bi,
                                                        (short)0, OrT, false, false);
            OiT = __builtin_amdgcn_wmma_f32_16x16x4_f32(false, iwT[c], false, Br,
                                                        (short)0, OiT, false, false);
            OiT = __builtin_amdgcn_wmma_f32_16x16x4_f32(false, rwT[c], false, Bi,
                                                        (short)0, OiT, false, false);
        }

        // ---- magnitude with folded 1/65536^2 scale; stage transposed tile in LDS
        const float s = 1.0f / 4294967296.0f;   // 2^-32
#pragma unroll
        for (int v = 0; v < 8; ++v) {
            const int m = v + hi * 8;
            const float a = OrT[v] * s;
            const float b = OiT[v] * s;
            ldsMg[m * LDS_STRIDE + col] = sqrtf(a * a + b * b);
        }
        __syncthreads();  // RAW: magnitudes visible across lanes

        // ---- coalesced row-major writeback: mag[r][c] = ldsMg[c*17 + r]
        // 4 full 128B bursts + one half-wave tail, all under a single active guard.
        float* om = out + (size_t)matC * MATF;
        float vout[5];
#pragma unroll
        for (int u = 0; u < 4; ++u) {
            const int t = lane + 32 * u;                 // 0..127 < 144
            const int r = t / IMG, cc = t - r * IMG;
            vout[u] = ldsMg[cc * LDS_STRIDE + r];
        }
        {
            const int t = 128 + lane;                    // 128..159, valid if < 144
            const int tc = t < MATF ? t : (MATF - 1);
            const int r = tc / IMG, cc = tc - r * IMG;
            vout[4] = ldsMg[cc * LDS_STRIDE + r];
        }
        if (active) {
#pragma unroll
            for (int u = 0; u < 4; ++u) om[lane + 32 * u] = vout[u];
            if (lane < 16) om[128 + lane] = vout[4];
        }
    }
}

extern "C" void kernel_launch(void* const* d_in, const int* in_sizes, int n_in,
                              void* d_out, int out_size, void* d_ws, size_t ws_size,
                              hipStream_t stream) {
    (void)n_in; (void)out_size; (void)d_ws; (void)ws_size;
    const float* x  = (const float*)d_in[0];
    const float* wr = (const float*)d_in[1];
    const float* wi = (const float*)d_in[2];
    float* out = (float*)d_out;

    const int nMat = in_sizes[0] / MATF;     // 100000 for the reference shapes
    if (nMat <= 0) return;
    int blocks = (nMat + WAVES_PER_BLOCK - 1) / WAVES_PER_BLOCK;  // one matrix per wave
    dft12_wmma_kernel<<<blocks, BLOCK_THREADS, 0, stream>>>(x, wr, wi, out, nMat);
}